// ThermalFeatureExtractor_74345883894588
// MI455X (gfx1250) — compile-verified
//
#include <hip/hip_runtime.h>
#include <math.h>

#define NBINS   32
#define NPIX    262144      // 512*512
#define NBATCH  64
#define ACC_PER_B 34        // sum, sumsq, hist[32]
#define K1PAD   36          // 34 padded to multiple of 4 for WMMA K
#define HID     256
#define EMB     128

typedef __attribute__((ext_vector_type(2))) float v2f;
typedef __attribute__((ext_vector_type(8))) float v8f;

// exp(-d^2/(2*sigma^2)) = exp2(d^2 * CEXP2); sigma = 5/32 -> 1/(2s^2) = 20.48
#define CEXP2 (-29.54639443740597f)   // -20.48 * log2(e)

#if __has_builtin(__builtin_amdgcn_exp2f)
#define EXP2F(x) __builtin_amdgcn_exp2f(x)
#else
#define EXP2F(x) exp2f(x)
#endif

// ---------------- phase 0: zero the per-batch accumulators ----------------
__global__ void tfe_zero_acc(float* __restrict__ acc) {
    int i = blockIdx.x * blockDim.x + threadIdx.x;
    if (i < NBATCH * ACC_PER_B) acc[i] = 0.0f;
}

// ---------------- phase 1: moments + soft gaussian histogram ----------------
__device__ __forceinline__ void tfe_px(float x, float& s, float& sq, float* h) {
    s += x;
    sq = fmaf(x, x, sq);
#pragma unroll
    for (int k = 0; k < NBINS; ++k) {
        const float ck = -1.0f + (float)k * (2.0f / 31.0f);
        float d = x - ck;
        h[k] += EXP2F(d * d * CEXP2);
    }
}

__global__ void __launch_bounds__(256)
tfe_hist(const float* __restrict__ ir, float* __restrict__ acc) {
    const int b   = blockIdx.y;
    const int tid = threadIdx.x;
    // each block: 4096 pixels; each thread: 4x float4 = 16 pixels, coalesced
    const float4* src =
        (const float4*)(ir + (size_t)b * NPIX + (size_t)blockIdx.x * 4096);

    float s = 0.0f, sq = 0.0f, h[NBINS];
#pragma unroll
    for (int k = 0; k < NBINS; ++k) h[k] = 0.0f;

#pragma unroll
    for (int i = 0; i < 4; ++i) {
        float4 p = src[i * 256 + tid];
        tfe_px(p.x, s, sq, h);
        tfe_px(p.y, s, sq, h);
        tfe_px(p.z, s, sq, h);
        tfe_px(p.w, s, sq, h);
    }

    // wave32 butterfly reduction (warpSize == 32 on gfx1250)
#pragma unroll
    for (int off = 16; off > 0; off >>= 1) {
        s  += __shfl_xor(s,  off, 32);
        sq += __shfl_xor(sq, off, 32);
#pragma unroll
        for (int k = 0; k < NBINS; ++k) h[k] += __shfl_xor(h[k], off, 32);
    }

    __shared__ float sacc[ACC_PER_B];
    if (tid < ACC_PER_B) sacc[tid] = 0.0f;
    __syncthreads();
    if ((tid & 31) == 0) {               // one lane per wave -> LDS f32 atomics
        atomicAdd(&sacc[0], s);
        atomicAdd(&sacc[1], sq);
#pragma unroll
        for (int k = 0; k < NBINS; ++k) atomicAdd(&sacc[2 + k], h[k]);
    }
    __syncthreads();
    if (tid < ACC_PER_B) atomicAdd(&acc[b * ACC_PER_B + tid], sacc[tid]);
}

// ---------------- phase 2: finalize stats -> padded (64 x 36) stats_vec ----
__global__ void tfe_finalize(const float* __restrict__ acc,
                             float* __restrict__ apad) {
    int b = threadIdx.x;
    if (b >= NBATCH) return;
    const float* a = acc + b * ACC_PER_B;
    const float n = (float)NPIX;
    float s = a[0], sq = a[1];
    float mean = s / n;
    float var  = (sq - s * s / n) / (n - 1.0f);     // ddof = 1
    float sd   = fmaxf(sqrtf(fmaxf(var, 0.0f)), 1e-6f);

    float hsum = 0.0f, h[NBINS];
#pragma unroll
    for (int k = 0; k < NBINS; ++k) { h[k] = a[2 + k] / n; hsum += h[k]; }
    float inv = 1.0f / (hsum + 1e-8f);

    float* o = apad + b * K1PAD;
    o[0] = mean; o[1] = sd;
#pragma unroll
    for (int k = 0; k < NBINS; ++k) o[2 + k] = h[k] * inv;
    o[34] = 0.0f; o[35] = 0.0f;                     // K padding for WMMA
}

// ---------------- phase 3: MLP with V_WMMA_F32_16X16X4_F32 -----------------
__global__ void __launch_bounds__(256)
tfe_mlp(const float* __restrict__ apad,
        const float* __restrict__ W1, const float* __restrict__ b1,
        const float* __restrict__ W2, const float* __restrict__ b2,
        float* __restrict__ out) {
    __shared__ float hbuf[NBATCH * HID];            // 64 KB hidden activations
    const int tid  = threadIdx.x;
    const int wave = tid >> 5;
    const int lane = tid & 31;
    const int lh   = lane & 15;
    const int hi2  = (lane >= 16) ? 2 : 0;          // K sub-offset (A/B frags)
    const int hi8  = (lane >= 16) ? 8 : 0;          // M sub-offset (C/D frags)

    // ---- GEMM1: (64x36) @ (34x256) + b1, exact GELU -> hbuf ----
    for (int t = wave; t < 64; t += 8) {            // 4x16 tiles of 16x16
        const int tm = t >> 4, tn = t & 15;
        const int m  = tm * 16 + lh;
        const int nn = tn * 16 + lh;
        v8f acc = {};
#pragma unroll
        for (int k0 = 0; k0 < 9; ++k0) {            // K = 36 in steps of 4
            const int kb = k0 * 4 + hi2;
            v2f av = *(const v2f*)(apad + m * K1PAD + kb);
            v2f bv;
            bv.x = (kb     < 34) ? W1[kb * HID + nn]       : 0.0f;
            bv.y = (kb + 1 < 34) ? W1[(kb + 1) * HID + nn] : 0.0f;
            acc = __builtin_amdgcn_wmma_f32_16x16x4_f32(
                false, av, false, bv, (short)0, acc, false, false);
        }
        const float bias = b1[nn];
#pragma unroll
        for (int v = 0; v < 8; ++v) {
            const int row = tm * 16 + v + hi8;
            float x = acc[v] + bias;
            float g = 0.5f * x * (1.0f + erff(x * 0.70710678118654752f));
            hbuf[row * HID + nn] = g;
        }
    }
    __syncthreads();

    // ---- GEMM2: (64x256) @ (256x128) + b2 -> out ----
    for (int t = wave; t < 32; t += 8) {            // 4x8 tiles of 16x16
        const int tm = t >> 3, tn = t & 7;
        const int m  = tm * 16 + lh;
        const int nn = tn * 16 + lh;
        v8f acc = {};
#pragma unroll 8
        for (int k0 = 0; k0 < 64; ++k0) {           // K = 256 in steps of 4
            const int kb = k0 * 4 + hi2;
            v2f av = *(const v2f*)(hbuf + m * HID + kb);
            v2f bv;
            bv.x = W2[kb * EMB + nn];
            bv.y = W2[(kb + 1) * EMB + nn];
            acc = __builtin_amdgcn_wmma_f32_16x16x4_f32(
                false, av, false, bv, (short)0, acc, false, false);
        }
        const float bias = b2[nn];
#pragma unroll
        for (int v = 0; v < 8; ++v) {
            const int row = tm * 16 + v + hi8;
            out[row * EMB + nn] = acc[v] + bias;
        }
    }
}

// ---------------------------------------------------------------------------
extern "C" void kernel_launch(void* const* d_in, const int* in_sizes, int n_in,
                              void* d_out, int out_size, void* d_ws, size_t ws_size,
                              hipStream_t stream) {
    const float* ir = (const float*)d_in[0];
    const float* W1 = (const float*)d_in[1];
    const float* b1 = (const float*)d_in[2];
    const float* W2 = (const float*)d_in[3];
    const float* b2 = (const float*)d_in[4];
    // d_in[5] = bin_centers (recomputed analytically on device)

    float* acc  = (float*)d_ws;                 // 64*34 accumulators
    float* apad = acc + NBATCH * ACC_PER_B;     // 64*36 padded stats_vec
    float* out  = (float*)d_out;                // 64*128 f32

    tfe_zero_acc<<<dim3((NBATCH * ACC_PER_B + 255) / 256), dim3(256), 0, stream>>>(acc);
    tfe_hist    <<<dim3(NPIX / 4096, NBATCH),              dim3(256), 0, stream>>>(ir, acc);
    tfe_finalize<<<dim3(1),                                dim3(64),  0, stream>>>(acc, apad);
    tfe_mlp     <<<dim3(1),                                dim3(256), 0, stream>>>(apad, W1, b1, W2, b2, out);
}